// CTCDecoder_42064909697277
// MI455X (gfx1250) — compile-verified
//
#include <hip/hip_runtime.h>

// ---------------------------------------------------------------------------
// CTC greedy decode for MI455X (gfx1250, wave32).
//   Phase 1: per-(b,t) argmax over 128 classes. One wave per row:
//            32 lanes x float4 = 128 contiguous floats (512B, fully coalesced).
//            Rows are streamed global->LDS with CDNA5 async copies
//            (double buffered, ASYNCcnt-tracked).
//   Phase 2: per-row collapse (merge repeats, drop blank=C-1), block scan,
//            scatter into -1-filled row. d_out doubles as the intermediate.
// ---------------------------------------------------------------------------

#ifndef __has_builtin
#define __has_builtin(x) 0
#endif

#if __has_builtin(__builtin_amdgcn_global_load_async_to_lds_b128)
#define CTC_ASYNC_LDS 1
#else
#define CTC_ASYNC_LDS 0
#endif

#if __has_builtin(__builtin_amdgcn_s_wait_asynccnt)
#define CTC_WAIT_ASYNC(n) __builtin_amdgcn_s_wait_asynccnt(n)
#else
#define CTC_WAIT_ASYNC(n) asm volatile("s_wait_asynccnt %0" ::"i"(n) : "memory")
#endif

// The builtin's parameter type (per hipcc diagnostic) is a pointer to a
// 128-bit int vector: `int __attribute__((vector_size(16)))`.
typedef int ctc_v4i __attribute__((vector_size(16)));
typedef __attribute__((address_space(1))) ctc_v4i ctc_g_v4i;  // global
typedef __attribute__((address_space(3))) ctc_v4i ctc_l_v4i;  // LDS

#if CTC_ASYNC_LDS
__device__ __forceinline__ void ctc_async_copy16(const float* g, float* l) {
  __builtin_amdgcn_global_load_async_to_lds_b128(
      (ctc_g_v4i*)(const_cast<float*>(g)),
      (ctc_l_v4i*)l,
      /*offset=*/0, /*cpol=*/0);
}
#endif

constexpr int kC     = 128;  // classes
constexpr int kT     = 1024; // timesteps
constexpr int kWaves = 8;    // waves per block (256 threads, wave32)
constexpr int kIters = 4;    // rows per wave (enables double buffering)

// ---------------------------------------------------------------------------
// Phase 1: argmax over class axis. am[row] = argmax_c logits[row, c]
// ---------------------------------------------------------------------------
__global__ __launch_bounds__(256) void ctc_argmax_kernel(
    const float* __restrict__ logits, int* __restrict__ am, int nrows) {
  const int lane = threadIdx.x & 31;
  const int wave = threadIdx.x >> 5;
  const long long rowBase =
      (long long)(blockIdx.x * kWaves + wave) * (long long)kIters;

#if CTC_ASYNC_LDS
  __shared__ alignas(16) float buf[kWaves][2][kC];
  float* lbase = &buf[wave][0][0];
  const float* gbase = logits + rowBase * kC;
  if (rowBase < nrows) {
    ctc_async_copy16(gbase + lane * 4, lbase + lane * 4);  // prime buffer 0
  }
#endif

  for (int i = 0; i < kIters; ++i) {
    const long long row = rowBase + i;
    if (row >= nrows) break;

    float4 v;
#if CTC_ASYNC_LDS
    if (i + 1 < kIters && row + 1 < nrows) {
      // Prefetch next row into the other buffer, wait only for the older copy.
      ctc_async_copy16(gbase + (long long)(i + 1) * kC + lane * 4,
                       lbase + ((i + 1) & 1) * kC + lane * 4);
      CTC_WAIT_ASYNC(1);
    } else {
      CTC_WAIT_ASYNC(0);
    }
    v = *(const float4*)(lbase + (i & 1) * kC + lane * 4);
#else
    v = *(const float4*)(logits + row * kC + lane * 4);
#endif

    // Lane-local argmax of 4 (first-index wins on ties, like jnp.argmax).
    float bv = v.x;
    int bi = lane * 4;
    if (v.y > bv) { bv = v.y; bi = lane * 4 + 1; }
    if (v.z > bv) { bv = v.z; bi = lane * 4 + 2; }
    if (v.w > bv) { bv = v.w; bi = lane * 4 + 3; }

    // wave32 butterfly reduction, min-index tie-break (associative).
#pragma unroll
    for (int m = 16; m >= 1; m >>= 1) {
      float ov = __shfl_xor(bv, m, 32);
      int   oi = __shfl_xor(bi, m, 32);
      if (ov > bv || (ov == bv && oi < bi)) { bv = ov; bi = oi; }
    }

    if (lane == 0) am[row] = bi;
  }
}

// ---------------------------------------------------------------------------
// Phase 2: per-row collapse + compact scatter. `out` row holds argmax ints on
// entry (written by phase 1) and the final decode (-1 padded) on exit.
// ---------------------------------------------------------------------------
__global__ __launch_bounds__(256) void ctc_collapse_kernel(
    int* __restrict__ out, int blank) {
  const int b = blockIdx.x;
  int* row = out + (long long)b * kT;
  const int tid = threadIdx.x;  // 0..255, 4 timesteps each

  __shared__ int am[kT];
  __shared__ int psum[256];

  // Stage the row (argmax indices) in LDS.
  const int4 v = ((const int4*)row)[tid];
  am[4 * tid + 0] = v.x;
  am[4 * tid + 1] = v.y;
  am[4 * tid + 2] = v.z;
  am[4 * tid + 3] = v.w;
  __syncthreads();

  // keep[t] = (am[t] != am[t-1]) && (am[t] != blank), am[-1] = -1.
  const int t0 = 4 * tid;
  int prev = (t0 == 0) ? -1 : am[t0 - 1];
  int a[4], k[4], loc = 0;
#pragma unroll
  for (int j = 0; j < 4; ++j) {
    a[j] = am[t0 + j];
    k[j] = (a[j] != prev && a[j] != blank) ? 1 : 0;
    prev = a[j];
    loc += k[j];
  }
  psum[tid] = loc;
  __syncthreads();

  // Hillis-Steele inclusive scan over the 256 thread partials.
#pragma unroll
  for (int off = 1; off < 256; off <<= 1) {
    const int val = psum[tid];
    const int add = (tid >= off) ? psum[tid - off] : 0;
    __syncthreads();
    psum[tid] = val + add;
    __syncthreads();
  }
  int pos = (tid == 0) ? 0 : psum[tid - 1];  // exclusive prefix for this thread

  // Reset the row to -1, then scatter kept tokens to compacted columns.
  ((int4*)row)[tid] = make_int4(-1, -1, -1, -1);
  __syncthreads();
#pragma unroll
  for (int j = 0; j < 4; ++j) {
    if (k[j]) { row[pos] = a[j]; ++pos; }
  }
}

// ---------------------------------------------------------------------------
// Host-side launch.
// ---------------------------------------------------------------------------
extern "C" void kernel_launch(void* const* d_in, const int* in_sizes, int n_in,
                              void* d_out, int out_size, void* d_ws,
                              size_t ws_size, hipStream_t stream) {
  (void)n_in; (void)d_ws; (void)ws_size; (void)out_size;

  const float* logits = (const float*)d_in[0];
  int* out = (int*)d_out;  // reference output dtype is int32

  const int C = 128;
  const int T = kT;
  const int rows = in_sizes[0] / C;  // B*T = 262144
  const int B = rows / T;            // 256
  const int blank = C - 1;           // 127

  const int rowsPerBlock = kWaves * kIters;  // 32
  const dim3 grid1((rows + rowsPerBlock - 1) / rowsPerBlock);
  hipLaunchKernelGGL(ctc_argmax_kernel, grid1, dim3(256), 0, stream,
                     logits, out, rows);
  hipLaunchKernelGGL(ctc_collapse_kernel, dim3(B), dim3(256), 0, stream,
                     out, blank);
}